// CrossCCC_17557826306350
// MI455X (gfx1250) — compile-verified
//
#include <hip/hip_runtime.h>

// ---------------------------------------------------------------------------
// CrossCCC on MI455X (gfx1250, wave32).
//
// ws layout (floats):
//   [0]   sum_p      [1] sumsq_p    [2] sum_gt    [3] sumsq_gt   (atomic accum)
//   [4..259]   R[256]  : cross-correlation tile D[q][r] = R(16q + r)
//   [260..509] tail1[n]: sum of last n elements of p
//   [510..759] tail2[n]: sum of squares of last n elements of p
// ---------------------------------------------------------------------------

typedef __attribute__((ext_vector_type(2))) float v2f;
typedef __attribute__((ext_vector_type(8))) float v8f;

#define NLAGS 250

__global__ void ccc_zero_ws(float* ws) {
    // zero the atomically-accumulated region ws[0..260)
    for (int i = threadIdx.x; i < 260; i += blockDim.x) ws[i] = 0.0f;
}

// 4-way grid-stride reduction: sum/sumsq of both arrays.
__global__ void ccc_reduce4(const float* __restrict__ p, const float* __restrict__ g,
                            float* __restrict__ ws, int T) {
    float sp = 0.f, sp2 = 0.f, sg = 0.f, sg2 = 0.f;
    for (int i = blockIdx.x * blockDim.x + threadIdx.x; i < T;
         i += gridDim.x * blockDim.x) {
        float a = p[i], b = g[i];
        sp += a; sp2 += a * a; sg += b; sg2 += b * b;
    }
    __shared__ float s0[256], s1[256], s2[256], s3[256];
    int t = threadIdx.x;
    s0[t] = sp; s1[t] = sp2; s2[t] = sg; s3[t] = sg2;
    __syncthreads();
    for (int s = 128; s > 0; s >>= 1) {
        if (t < s) { s0[t] += s0[t+s]; s1[t] += s1[t+s]; s2[t] += s2[t+s]; s3[t] += s3[t+s]; }
        __syncthreads();
    }
    if (t == 0) {
        atomicAdd(&ws[0], s0[0]); atomicAdd(&ws[1], s1[0]);
        atomicAdd(&ws[2], s2[0]); atomicAdd(&ws[3], s3[0]);
    }
}

// Tail partial sums over the last n elements of p (n < 250): tiny, one block.
__global__ void ccc_tails(const float* __restrict__ p, float* __restrict__ ws, int T) {
    int n = threadIdx.x;
    if (n >= NLAGS) return;
    float t1 = 0.f, t2 = 0.f;
    for (int j = T - n; j < T; ++j) { float a = p[j]; t1 += a; t2 += a * a; }
    ws[260 + n] = t1;
    ws[510 + n] = t2;
}

// Cross-correlation via fp32 WMMA on the Toeplitz factorization:
//   D[q][r] = sum_k p[k - 16q] * g[k + r],  lag n = 16q + r.
// Each wave accumulates one 16x16 f32 tile over its k-chunk, K=4 per WMMA.
// Interior waves (bounds provably safe, wave-uniform test) run an unguarded
// 2x-unrolled loop: 3 loads + 1 wmma per K=4. Boundary waves take the
// guarded path.
__global__ void ccc_xcorr_wmma(const float* __restrict__ p, const float* __restrict__ g,
                               float* __restrict__ R, int T, int kTotal, int chunk) {
    int wave = (blockIdx.x * blockDim.x + threadIdx.x) >> 5;
    int lane = threadIdx.x & 31;
    int half = lane >> 4;    // 0: lanes 0-15, 1: lanes 16-31
    int l16  = lane & 15;

    int k0   = wave * chunk;
    int kEnd = k0 + chunk;
    if (kEnd > kTotal) kEnd = kTotal;

    // A (16x4 f32): lane group holds K = 2*half + {0,1} for row M = l16
    //   A[M][kk] = p[k + kk - 16*M]   -> even base index, 8B-aligned pair
    const int aOff = 2 * half - 16 * l16;
    // B (4x16 f32): lane group holds rows K = 2*half + {0,1}, column r = l16
    //   B[kk][r] = g[k + kk + r]      -> odd for odd lanes, scalar loads
    const int bOff = 2 * half + l16;

    v8f acc = {};  // fp32 accumulator tile (8 VGPRs, full 16x16 across wave)

    // Wave-uniform interior test:
    //   min p index = k0 + aOff >= k0 - 240
    //   max p index = (kEnd-1) + aOff + 1 <= kEnd + 3
    //   max g index = (kEnd-1) + bOff + 1 <= kEnd + 17
    const bool interior = (k0 >= 240) && (kEnd + 18 <= T);

    int k = k0;
    if (interior) {
        // fast path: no bounds checks, 2 WMMAs per iteration
        for (; k + 8 <= kEnd; k += 8) {
            v2f a0 = *(const v2f*)(p + (k + aOff));
            v2f b0; b0.x = g[k + bOff]; b0.y = g[k + bOff + 1];
            acc = __builtin_amdgcn_wmma_f32_16x16x4_f32(
                false, a0, false, b0, (short)0, acc, false, false);
            v2f a1 = *(const v2f*)(p + (k + 4 + aOff));
            v2f b1; b1.x = g[k + 4 + bOff]; b1.y = g[k + 4 + bOff + 1];
            acc = __builtin_amdgcn_wmma_f32_16x16x4_f32(
                false, a1, false, b1, (short)0, acc, false, false);
        }
        for (; k < kEnd; k += 4) {
            v2f a; a.x = p[k + aOff]; a.y = p[k + aOff + 1];
            v2f b; b.x = g[k + bOff]; b.y = g[k + bOff + 1];
            acc = __builtin_amdgcn_wmma_f32_16x16x4_f32(
                false, a, false, b, (short)0, acc, false, false);
        }
    } else {
        // boundary path: per-element guards (zero padding outside [0,T))
        for (; k < kEnd; k += 4) {
            v2f a, b;
            int pi0 = k + aOff;
            int pi1 = pi0 + 1;
            a.x = (pi0 >= 0 && pi0 < T) ? p[pi0] : 0.0f;
            a.y = (pi1 >= 0 && pi1 < T) ? p[pi1] : 0.0f;
            int gi0 = k + bOff;          // always >= 0
            int gi1 = gi0 + 1;
            b.x = (gi0 < T) ? g[gi0] : 0.0f;
            b.y = (gi1 < T) ? g[gi1] : 0.0f;
            acc = __builtin_amdgcn_wmma_f32_16x16x4_f32(
                false, a, false, b, (short)0, acc, false, false);
        }
    }

    // D layout: VGPR v -> M = v (lanes 0-15) or v+8 (lanes 16-31), N = l16.
    #pragma unroll
    for (int v = 0; v < 8; ++v) {
        int q = v + 8 * half;
        atomicAdd(&R[q * 16 + l16], acc[v]);
    }
}

// Final CCC formula + mean over 250 lags. One block of 256 threads.
__global__ void ccc_finalize(const float* __restrict__ ws, float* __restrict__ out, int T) {
    __shared__ float red[256];
    int n = threadIdx.x;
    float Tf       = (float)T;
    float sum_p    = ws[0], sumsq_p  = ws[1];
    float sum_gt   = ws[2], sumsq_gt = ws[3];
    float mean_gt  = sum_gt / Tf;
    float var_gt   = (sumsq_gt - Tf * mean_gt * mean_gt) / (Tf - 1.0f);

    float ccc = 0.0f;
    if (n < NLAGS) {
        float S1        = sum_p   - ws[260 + n];
        float S2        = sumsq_p - ws[510 + n];
        float mean_pred = S1 / Tf;
        float var_pred  = (S2 - Tf * mean_pred * mean_pred) / (Tf - 1.0f);
        float Rn        = ws[4 + n];                 // sum p[j]*g[j+n]
        float cov       = (Rn - mean_gt * S1) / Tf;  // population covariance
        float dm        = mean_gt - mean_pred;
        float denom     = var_gt + var_pred + dm * dm;
        ccc = 2.0f * cov / denom;
    }
    red[n] = ccc;
    __syncthreads();
    for (int s = 128; s > 0; s >>= 1) {
        if (n < s) red[n] += red[n + s];
        __syncthreads();
    }
    if (n == 0) out[0] = 1.0f - red[0] / (float)NLAGS;
}

extern "C" void kernel_launch(void* const* d_in, const int* in_sizes, int n_in,
                              void* d_out, int out_size, void* d_ws, size_t ws_size,
                              hipStream_t stream) {
    const float* p = (const float*)d_in[0];   // prediction
    const float* g = (const float*)d_in[1];   // ground_truth
    float* out = (float*)d_out;
    float* ws  = (float*)d_ws;
    int T = in_sizes[0];

    ccc_zero_ws<<<1, 256, 0, stream>>>(ws);
    ccc_reduce4<<<512, 256, 0, stream>>>(p, g, ws, T);
    ccc_tails<<<1, 256, 0, stream>>>(p, ws, T);

    // k ranges over [0, T + 240): max useful k is (T-1) + 16*15.
    int kTotal = T + 240;
    int nBlocks = 512;
    int waves = nBlocks * (256 / 32);
    int chunk = ((kTotal + waves - 1) / waves + 3) & ~3;   // multiple of 4 (K step)
    ccc_xcorr_wmma<<<nBlocks, 256, 0, stream>>>(p, g, ws + 4, T, kTotal, chunk);

    ccc_finalize<<<1, 256, 0, stream>>>(ws, out, T);
}